// HyperbolicContrastiveLoss_16707422781501
// MI455X (gfx1250) — compile-verified
//
#include <hip/hip_runtime.h>

typedef __attribute__((ext_vector_type(2))) float v2f;
typedef __attribute__((ext_vector_type(8))) float v8f;
typedef int v4i_vs __attribute__((vector_size(16)));   // matches builtin's pointee type

#define NROWS 4096
#define DDIM  256
#define JSLICES 8
#define BM 32        // rows per block (2 M-subtiles)
#define BN 64        // cols per j-tile (4 N-subtiles)
#define KCH 64       // K staged per LDS chunk
#define NKC (DDIM / KCH)                 // 4 k-chunks per j-tile
#define NJT ((NROWS / JSLICES) / BN)     // 8 j-tiles per slice
#define NT  (NJT * NKC)                  // 32 pipelined chunks
#define AS 260       // LDS stride (floats) for A tile: 260%64==4 -> conflict-free column reads
#define BS 68        // LDS stride (floats) for B chunk: 68%64==4 -> conflict-free

#if defined(__has_builtin)
#if __has_builtin(__builtin_amdgcn_global_load_async_to_lds_b128) && \
    __has_builtin(__builtin_amdgcn_s_wait_asynccnt)
#define HAS_ASYNC_LDS 1
#endif
#endif

// 16-byte per-lane copy global -> LDS.
__device__ __forceinline__ void cp16_to_lds(float* lds_dst, const float* gsrc) {
#ifdef HAS_ASYNC_LDS
  __builtin_amdgcn_global_load_async_to_lds_b128(
      (__attribute__((address_space(1))) v4i_vs*)(gsrc),
      (__attribute__((address_space(3))) v4i_vs*)(lds_dst), 0, 0);
#else
  *(float4*)lds_dst = *(const float4*)gsrc;
#endif
}

__device__ __forceinline__ void wait_async_le4() {
#ifdef HAS_ASYNC_LDS
  __builtin_amdgcn_s_wait_asynccnt(4);
#endif
}
__device__ __forceinline__ void wait_async_zero() {
#ifdef HAS_ASYNC_LDS
  __builtin_amdgcn_s_wait_asynccnt(0);
#endif
}

// ---------------- Pass 1: Poincare-ball projection + row stats ----------------
__global__ __launch_bounds__(256) void hyp_proj_kernel(const float* __restrict__ x,
                                                       float* __restrict__ xp,
                                                       float* __restrict__ xs,
                                                       float* __restrict__ inv1m) {
  int wave = threadIdx.x >> 5;
  int lane = threadIdx.x & 31;
  int row  = blockIdx.x * 8 + wave;            // one wave32 per row
  const float4* xr = (const float4*)(x + row * DDIM);
  float4 v0 = xr[lane];
  float4 v1 = xr[lane + 32];
  float ss = v0.x*v0.x + v0.y*v0.y + v0.z*v0.z + v0.w*v0.w
           + v1.x*v1.x + v1.y*v1.y + v1.z*v1.z + v1.w*v1.w;
  for (int m = 1; m < 32; m <<= 1) ss += __shfl_xor(ss, m, 32);
  float norm = fmaxf(sqrtf(ss), 1e-5f);
  const float max_norm = 1.0f - 1e-5f;          // c == 1
  float scale = (norm > max_norm) ? (max_norm / norm) : 1.0f;
  float4* xpr = (float4*)(xp + row * DDIM);
  v0.x *= scale; v0.y *= scale; v0.z *= scale; v0.w *= scale;
  v1.x *= scale; v1.y *= scale; v1.z *= scale; v1.w *= scale;
  xpr[lane]      = v0;
  xpr[lane + 32] = v1;
  if (lane == 0) {
    float s2 = ss * scale * scale;              // ||x_proj||^2
    xs[row] = s2;
    inv1m[row] = 1.0f / (1.0f - s2);            // 1-s2 > 0 always after projection
  }
}

// ---------------- Pass 2: fp32 WMMA gram + fused hyperbolic epilogue ----------------
__global__ __launch_bounds__(256) void hyp_gram_loss_kernel(
    const float* __restrict__ xp, const float* __restrict__ xs,
    const float* __restrict__ inv1m, const int* __restrict__ labels,
    float* __restrict__ den_part, float* __restrict__ num_part,
    float* __restrict__ cnt_part) {
  __shared__ __align__(16) float a_sh[BM * AS];
  __shared__ __align__(16) float b_sh[2][BN * BS];   // double-buffered B chunks
  __shared__ float red_sh[3][8][16];

  const int tid  = threadIdx.x;
  const int wave = tid >> 5, lane = tid & 31;
  const int mi = wave >> 2, ni = wave & 3;       // 2x4 wave grid over the 32x64 tile
  const int half = lane >> 4, lr = lane & 15;
  const int mb = blockIdx.x * BM;
  const int slice = blockIdx.y;
  const int jspan = NROWS / JSLICES;             // 512 columns per slice

  // Stage the whole A block (BM x D) once via async copies (8 x b128 per thread).
  #pragma unroll
  for (int i = 0; i < (BM * DDIM / 4) / 256; ++i) {
    int f4 = i * 256 + tid;
    int r  = f4 >> 6;                            // D/4 == 64
    int c4 = f4 & 63;
    cp16_to_lds(&a_sh[r * AS + c4 * 4], xp + (mb + r) * DDIM + c4 * 4);
  }
  // Issue first B chunk (j-tile 0, kc 0) into buffer 0.
  {
    int jb = slice * jspan;
    #pragma unroll
    for (int i = 0; i < (BN * KCH / 4) / 256; ++i) {
      int f4 = i * 256 + tid;
      int r  = f4 >> 4;                          // KCH/4 == 16
      int c4 = f4 & 15;
      cp16_to_lds(&b_sh[0][r * BS + c4 * 4], xp + (jb + r) * DDIM + c4 * 4);
    }
  }

  // Per-wave row-constant state (M index = mb + mi*16 + 8*half + v, per C layout).
  int   mrow[8]; float xsm[8], ivm[8]; int lbm[8];
  #pragma unroll
  for (int v = 0; v < 8; ++v) {
    int mg = mb + mi * 16 + 8 * half + v;
    mrow[v] = mg; xsm[v] = xs[mg]; ivm[v] = inv1m[mg]; lbm[v] = labels[mg];
  }
  float den8[8], num8[8], cnt8[8];
  #pragma unroll
  for (int v = 0; v < 8; ++v) { den8[v] = 0.f; num8[v] = 0.f; cnt8[v] = 0.f; }

  v8f acc = {0.f, 0.f, 0.f, 0.f, 0.f, 0.f, 0.f, 0.f};

  // Flattened (j-tile x k-chunk) software pipeline over NT chunks.
  for (int t = 0; t < NT; ++t) {
    const int jt  = t >> 2;
    const int kci = t & 3;
    const int cur = t & 1;

    // Prefetch chunk t+1 into the idle buffer (freed by the previous end barrier).
    if (t + 1 < NT) {
      int tn  = t + 1;
      int jbn = slice * jspan + (tn >> 2) * BN;
      int kcn = (tn & 3) * KCH;
      #pragma unroll
      for (int i = 0; i < (BN * KCH / 4) / 256; ++i) {
        int f4 = i * 256 + tid;
        int r  = f4 >> 4;
        int c4 = f4 & 15;
        cp16_to_lds(&b_sh[tn & 1][r * BS + c4 * 4],
                    xp + (jbn + r) * DDIM + kcn + c4 * 4);
      }
      wait_async_le4();    // async loads retire in order: chunk t (and A) have landed
    } else {
      wait_async_zero();   // last chunk: drain everything
    }
    __syncthreads();

    if (kci == 0) {
      v8f z = {0.f, 0.f, 0.f, 0.f, 0.f, 0.f, 0.f, 0.f};
      acc = z;
    }
    const int kbase = kci * KCH;
    #pragma unroll
    for (int kk = 0; kk < KCH; kk += 4) {
      // A frag 16x4: lanes 0-15 hold K={k,k+1}, lanes 16-31 K={k+2,k+3} (row = lr)
      v2f a = *(const v2f*)&a_sh[(mi * 16 + lr) * AS + kbase + kk + 2 * half];
      // B frag 4x16: column n = lr; gram B[k][n] = Xp[jb+n][k]
      v2f b = *(const v2f*)&b_sh[cur][(ni * 16 + lr) * BS + kk + 2 * half];
      acc = __builtin_amdgcn_wmma_f32_16x16x4_f32(
          false, a, false, b, (short)0, acc, false, false);
    }

    if (kci == 3) {
      // Fused epilogue: exp(-2*atanh(s)/T) == ((1-s)/(1+s))^(1/T), 1/T == 10.
      const int jb = slice * jspan + jt * BN;
      const int jg = jb + ni * 16 + lr;          // column index for this lane (all 8 v's)
      const float xsn = xs[jg];
      const float ivn = inv1m[jg];
      const int   lbn = labels[jg];
      #pragma unroll
      for (int v = 0; v < 8; ++v) {
        float g    = acc[v];
        float diff = fmaxf(xsm[v] + xsn - 2.0f * g, 0.0f);
        float ratio = fminf(2.0f * diff * ivm[v] * ivn, 1.0f - 1e-5f);
        float s  = sqrtf(ratio);
        float r  = (1.0f - s) / (1.0f + s);
        float r2 = r * r, r4 = r2 * r2, r8 = r4 * r4;
        float e  = r8 * r2;                      // r^10 == exp(-d/T)
        bool self = (mrow[v] == jg);
        e = self ? 0.0f : e;
        den8[v] += e;
        bool same = (lbm[v] == lbn) && !self;
        num8[v] += same ? e : 0.0f;
        cnt8[v] += same ? 1.0f : 0.0f;
      }
    }
    __syncthreads();   // release buffer `cur` for the prefetch at iteration t+1
  }

  // Reduce the 16 column-lanes of each half; lanes 0 / 16 own the row results.
  #pragma unroll
  for (int v = 0; v < 8; ++v) {
    float d = den8[v], nu = num8[v], ct = cnt8[v];
    for (int m = 1; m < 16; m <<= 1) {
      d  += __shfl_xor(d,  m, 32);
      nu += __shfl_xor(nu, m, 32);
      ct += __shfl_xor(ct, m, 32);
    }
    if (lr == 0) {
      red_sh[0][wave][8 * half + v] = d;
      red_sh[1][wave][8 * half + v] = nu;
      red_sh[2][wave][8 * half + v] = ct;
    }
  }
  __syncthreads();

  // Combine the 4 N-waves that share each row block; one writer per (slice,row).
  if (tid < BM) {
    int r = tid, rmi = r >> 4, rr = r & 15;
    float d = 0.f, nu = 0.f, ct = 0.f;
    for (int w = rmi * 4; w < rmi * 4 + 4; ++w) {
      d += red_sh[0][w][rr]; nu += red_sh[1][w][rr]; ct += red_sh[2][w][rr];
    }
    int idx = slice * NROWS + mb + r;
    den_part[idx] = d; num_part[idx] = nu; cnt_part[idx] = ct;
  }
}

// ---------------- Pass 3: fold slices, per-row loss, scalar reduction ----------------
__global__ __launch_bounds__(256) void hyp_finalize_kernel(
    const float* __restrict__ den_part, const float* __restrict__ num_part,
    const float* __restrict__ cnt_part, float* __restrict__ out) {
  __shared__ float sl[256], sv[256];
  float lsum = 0.f, vsum = 0.f;
  for (int r = threadIdx.x; r < NROWS; r += 256) {
    float d = 0.f, nu = 0.f, ct = 0.f;
    for (int s = 0; s < JSLICES; ++s) {
      d += den_part[s * NROWS + r];
      nu += num_part[s * NROWS + r];
      ct += cnt_part[s * NROWS + r];
    }
    nu = fmaxf(nu, 1e-8f);
    d  = fmaxf(d, 1e-8f);
    float loss  = -logf(nu / (nu + d));
    float valid = (ct > 0.5f) ? 1.0f : 0.0f;
    lsum += loss * valid;
    vsum += valid;
  }
  sl[threadIdx.x] = lsum; sv[threadIdx.x] = vsum;
  __syncthreads();
  for (int st = 128; st > 0; st >>= 1) {
    if (threadIdx.x < st) {
      sl[threadIdx.x] += sl[threadIdx.x + st];
      sv[threadIdx.x] += sv[threadIdx.x + st];
    }
    __syncthreads();
  }
  if (threadIdx.x == 0)
    out[0] = (sv[0] > 0.0f) ? (sl[0] / fmaxf(sv[0], 1.0f)) : 0.0f;
}

extern "C" void kernel_launch(void* const* d_in, const int* in_sizes, int n_in,
                              void* d_out, int out_size, void* d_ws, size_t ws_size,
                              hipStream_t stream) {
  const float* emb    = (const float*)d_in[0];
  const int*   labels = (const int*)d_in[1];
  float* out = (float*)d_out;

  float* xp       = (float*)d_ws;                       // 4096*256
  float* xs       = xp + NROWS * DDIM;                  // 4096
  float* inv1m    = xs + NROWS;                         // 4096
  float* den_part = inv1m + NROWS;                      // 8*4096
  float* num_part = den_part + JSLICES * NROWS;         // 8*4096
  float* cnt_part = num_part + JSLICES * NROWS;         // 8*4096

  hyp_proj_kernel<<<NROWS / 8, 256, 0, stream>>>(emb, xp, xs, inv1m);
  dim3 grid(NROWS / BM, JSLICES);                       // 128 x 8 workgroups
  hyp_gram_loss_kernel<<<grid, 256, 0, stream>>>(xp, xs, inv1m, labels,
                                                 den_part, num_part, cnt_part);
  hyp_finalize_kernel<<<1, 256, 0, stream>>>(den_part, num_part, cnt_part, out);
}